// LightAggregator_79216376807725
// MI455X (gfx1250) — compile-verified
//
#include <hip/hip_runtime.h>
#include <hip/hip_bf16.h>

typedef float v2f __attribute__((ext_vector_type(2)));
typedef float v8f __attribute__((ext_vector_type(8)));

#define D_DIM 64

// ---------------- utility kernels ----------------

__global__ void fill_zero_u32(unsigned int* p, int n) {
  int i = blockIdx.x * blockDim.x + threadIdx.x;
  int stride = gridDim.x * blockDim.x;
  for (; i < n; i += stride) p[i] = 0u;
}

__global__ void copy_u32(unsigned int* dst, const unsigned int* src, int n) {
  int i = blockIdx.x * blockDim.x + threadIdx.x;
  int stride = gridDim.x * blockDim.x;
  for (; i < n; i += stride) dst[i] = src[i];
}

__global__ void hist_kernel(const int* __restrict__ idx, int nnz, int* counts) {
  int i = blockIdx.x * blockDim.x + threadIdx.x;
  int stride = gridDim.x * blockDim.x;
  for (; i < nnz; i += stride) atomicAdd(&counts[idx[i]], 1);
}

// Single-workgroup exclusive scan: offsets[0]=0, offsets[i+1]=sum(counts[0..i]).
__global__ __launch_bounds__(1024) void exclusive_scan_kernel(
    const int* __restrict__ counts, int n, int* __restrict__ offsets) {
  __shared__ int sh[1024];
  __shared__ int carry_sh;
  const int tid = threadIdx.x;
  if (tid == 0) carry_sh = 0;
  __syncthreads();
  for (int base = 0; base < n; base += 1024) {
    int i = base + tid;
    int v = (i < n) ? counts[i] : 0;
    sh[tid] = v;
    __syncthreads();
    for (int off = 1; off < 1024; off <<= 1) {
      int t = (tid >= off) ? sh[tid - off] : 0;
      __syncthreads();
      sh[tid] += t;
      __syncthreads();
    }
    int incl = sh[tid];
    int carry = carry_sh;
    if (i < n) offsets[i + 1] = carry + incl;
    __syncthreads();
    if (tid == 1023) carry_sh = carry + incl;
    __syncthreads();
  }
  if (tid == 0) offsets[0] = 0;
}

// Bucket edges by `key`.
__global__ void scatter_pairs_kernel(const int* __restrict__ key,
                                     const int* __restrict__ other,
                                     const float* __restrict__ vals, int nnz,
                                     int* cursor, int* __restrict__ out_other,
                                     float* __restrict__ out_val) {
  int i = blockIdx.x * blockDim.x + threadIdx.x;
  int stride = gridDim.x * blockDim.x;
  for (; i < nnz; i += stride) {
    int pos = atomicAdd(&cursor[key[i]], 1);
    out_other[pos] = other[i];
    out_val[pos] = vals[i];
  }
}

// ---------------- WMMA segment-sum SpMM ----------------
// One wave32 per group of 16 output segments. Segment-sorted edges are
// consumed 32 per outer iteration (coalesced per-lane loads, tail masked by
// cndmask — no scalar branches), then fed to V_WMMA_F32_16X16X4_F32 in 4-edge
// chunks: D(16x16) += A(16x4) x B(4x16), A[i][k] = val_k*(owner_k==i),
// B[k][j] = emb[src_k][tile*16+j]. Owner of each edge is computed once per
// 32 edges and redistributed with ds_bpermute. Full f32 precision; each
// output element written exactly once (no f32 atomics anywhere).
__global__ __launch_bounds__(32) void spmm_wmma16x16x4(
    const float* __restrict__ emb, const int* __restrict__ seg_off,
    const int* __restrict__ edge_src, const float* __restrict__ edge_val,
    int n_seg, float* __restrict__ out) {
  const int lane = threadIdx.x;
  const int seg0 = blockIdx.x * 16;
  const bool hi = lane >= 16;
  const int m = lane & 15;

  int bounds[17];
#pragma unroll
  for (int t = 0; t <= 16; ++t) {
    int s = seg0 + t;
    bounds[t] = seg_off[s < n_seg ? s : n_seg];
  }
  const int start = bounds[0];
  const int end = bounds[16];

  v8f acc0 = {}, acc1 = {}, acc2 = {}, acc3 = {};

  for (int p = start; p < end; p += 32) {  // uniform: EXEC stays all-ones
    // Per-lane coalesced load of 32 edges (tail lanes masked, not branched).
    const int e = p + lane;
    const bool ok = e < end;
    const float v32 = ok ? edge_val[e] : 0.0f;
    const int s32 = ok ? edge_src[e] : 0;
    // Owner of this lane's edge within the 16-segment group (once per 32
    // edges). Out-of-range edges get owner==16, which matches no row.
    int own32 = 0;
#pragma unroll
    for (int t = 0; t < 16; ++t) own32 += (e >= bounds[t + 1]) ? 1 : 0;

#pragma unroll
    for (int c = 0; c < 32; c += 4) {  // 8 chunks of K=4
      if (p + c >= end) break;         // uniform early-out
      // Lanes 0-15 need edges c,c+1 (K=0,1); lanes 16-31 need c+2,c+3.
      const int k0 = c + (hi ? 2 : 0);
      const float va = __shfl(v32, k0, 32);
      const float vb = __shfl(v32, k0 + 1, 32);
      const int oa = __shfl(own32, k0, 32);
      const int ob = __shfl(own32, k0 + 1, 32);
      const int ra = __shfl(s32, k0, 32);
      const int rb = __shfl(s32, k0 + 1, 32);

      v2f a;
      a.x = (oa == m) ? va : 0.0f;
      a.y = (ob == m) ? vb : 0.0f;

      const float* rowA = emb + (size_t)ra * D_DIM;
      const float* rowB = emb + (size_t)rb * D_DIM;
      v2f b0, b1, b2, b3;
      b0.x = rowA[m];      b0.y = rowB[m];
      b1.x = rowA[16 + m]; b1.y = rowB[16 + m];
      b2.x = rowA[32 + m]; b2.y = rowB[32 + m];
      b3.x = rowA[48 + m]; b3.y = rowB[48 + m];

      acc0 = __builtin_amdgcn_wmma_f32_16x16x4_f32(false, a, false, b0, (short)0, acc0, false, false);
      acc1 = __builtin_amdgcn_wmma_f32_16x16x4_f32(false, a, false, b1, (short)0, acc1, false, false);
      acc2 = __builtin_amdgcn_wmma_f32_16x16x4_f32(false, a, false, b2, (short)0, acc2, false, false);
      acc3 = __builtin_amdgcn_wmma_f32_16x16x4_f32(false, a, false, b3, (short)0, acc3, false, false);
    }
  }

  // C/D layout: VGPR r holds M=r (lanes 0-15) / M=r+8 (lanes 16-31), N=m.
  const int mbase = hi ? 8 : 0;
#pragma unroll
  for (int r = 0; r < 8; ++r) {
    int seg = seg0 + mbase + r;
    if (seg < n_seg) {
      float* orow = out + (size_t)seg * D_DIM + m;
      orow[0]  = acc0[r];
      orow[16] = acc1[r];
      orow[32] = acc2[r];
      orow[48] = acc3[r];
    }
  }
}

// ---------------- atomic fallback (if ws too small) ----------------

__global__ void atomic_fallback_kernel(const float* __restrict__ user_emb,
                                       const float* __restrict__ entity_emb,
                                       const int* __restrict__ rows,
                                       const int* __restrict__ cols,
                                       const float* __restrict__ vals, int nnz,
                                       float* entity_agg, float* user_agg) {
  unsigned int tid = blockIdx.x * blockDim.x + threadIdx.x;
  int e = (int)(tid >> 5);
  int part = (int)(tid & 31u);
  if (e >= nnz) return;
  int r = rows[e], c = cols[e];
  float v = vals[e];
  const float2* up = (const float2*)(user_emb + (size_t)r * D_DIM);
  const float2* ep = (const float2*)(entity_emb + (size_t)c * D_DIM);
  float2 u = up[part];
  float2 en = ep[part];
  atomicAdd(&entity_agg[(size_t)c * D_DIM + 2 * part + 0], v * u.x);
  atomicAdd(&entity_agg[(size_t)c * D_DIM + 2 * part + 1], v * u.y);
  atomicAdd(&user_agg[(size_t)r * D_DIM + 2 * part + 0], v * en.x);
  atomicAdd(&user_agg[(size_t)r * D_DIM + 2 * part + 1], v * en.y);
}

// ---------------- launch ----------------

extern "C" void kernel_launch(void* const* d_in, const int* in_sizes, int n_in,
                              void* d_out, int out_size, void* d_ws, size_t ws_size,
                              hipStream_t stream) {
  const float* user_emb = (const float*)d_in[0];
  const float* entity_emb = (const float*)d_in[1];
  const int* rows = (const int*)d_in[2];
  const int* cols = (const int*)d_in[3];
  const float* vals = (const float*)d_in[4];

  const int n_users = in_sizes[0] / D_DIM;
  const int n_entities = in_sizes[1] / D_DIM;
  const int nnz = in_sizes[2];

  float* entity_agg = (float*)d_out;
  float* user_agg = entity_agg + (size_t)n_entities * D_DIM;

  const size_t need =
      sizeof(int) * ((size_t)(n_entities + 1) + (n_users + 1) + n_entities +
                     n_users + 4 * (size_t)nnz);

  if (ws_size >= need) {
    int* off_col = (int*)d_ws;                  // n_entities + 1
    int* off_row = off_col + n_entities + 1;    // n_users + 1
    int* cur_col = off_row + n_users + 1;       // n_entities (also counts)
    int* cur_row = cur_col + n_entities;        // n_users   (also counts)
    int* erow_bycol = cur_row + n_users;        // nnz
    int* ecol_byrow = erow_bycol + nnz;         // nnz
    float* eval_bycol = (float*)(ecol_byrow + nnz);  // nnz
    float* eval_byrow = eval_bycol + nnz;            // nnz

    const int blk = 256;
    int gz = (n_entities + n_users + blk - 1) / blk;
    fill_zero_u32<<<gz, blk, 0, stream>>>((unsigned int*)cur_col,
                                          n_entities + n_users);

    int gh = (nnz + blk - 1) / blk;
    if (gh > 4096) gh = 4096;
    hist_kernel<<<gh, blk, 0, stream>>>(cols, nnz, cur_col);
    hist_kernel<<<gh, blk, 0, stream>>>(rows, nnz, cur_row);

    exclusive_scan_kernel<<<1, 1024, 0, stream>>>(cur_col, n_entities, off_col);
    exclusive_scan_kernel<<<1, 1024, 0, stream>>>(cur_row, n_users, off_row);

    copy_u32<<<(n_entities + blk - 1) / blk, blk, 0, stream>>>(
        (unsigned int*)cur_col, (const unsigned int*)off_col, n_entities);
    copy_u32<<<(n_users + blk - 1) / blk, blk, 0, stream>>>(
        (unsigned int*)cur_row, (const unsigned int*)off_row, n_users);

    scatter_pairs_kernel<<<gh, blk, 0, stream>>>(cols, rows, vals, nnz, cur_col,
                                                 erow_bycol, eval_bycol);
    scatter_pairs_kernel<<<gh, blk, 0, stream>>>(rows, cols, vals, nnz, cur_row,
                                                 ecol_byrow, eval_byrow);

    int gseg_e = (n_entities + 15) / 16;
    int gseg_u = (n_users + 15) / 16;
    spmm_wmma16x16x4<<<gseg_e, 32, 0, stream>>>(user_emb, off_col, erow_bycol,
                                                eval_bycol, n_entities,
                                                entity_agg);
    spmm_wmma16x16x4<<<gseg_u, 32, 0, stream>>>(entity_emb, off_row, ecol_byrow,
                                                eval_byrow, n_users, user_agg);
  } else {
    const int blk = 256;
    int nz = out_size;
    fill_zero_u32<<<(nz + blk - 1) / blk, blk, 0, stream>>>(
        (unsigned int*)d_out, nz);
    unsigned long long threads = (unsigned long long)nnz * 32ull;
    unsigned int grid = (unsigned int)((threads + blk - 1) / blk);
    atomic_fallback_kernel<<<grid, blk, 0, stream>>>(
        user_emb, entity_emb, rows, cols, vals, nnz, entity_agg, user_agg);
  }
}